// GGNN_47639777247408
// MI455X (gfx1250) — compile-verified
//
#include <hip/hip_runtime.h>
#include <hip/hip_bf16.h>

typedef __attribute__((ext_vector_type(2))) float v2f;
typedef __attribute__((ext_vector_type(8))) float v8f;

namespace {
constexpr int N    = 50000;
constexpr int H    = 128;
constexpr int H3   = 384;
constexpr int E    = 600000;
constexpr int NP   = 50016;            // padded to multiple of 32
constexpr int MBLK = N  / 16;          // 3125 row tiles (dense/max), exact
constexpr int GBLK = NP / 32;          // 1563 row tiles (GRU), exact on padded
}

// ---------------------------------------------------------------------------
// Scatter-add message passing: xn[dst] += h[src], one thread per (edge, 4 cols)
// ---------------------------------------------------------------------------
__global__ __launch_bounds__(256) void ggnn_scatter(
    const float* __restrict__ h, const int* __restrict__ src,
    const int* __restrict__ dst, float* __restrict__ xn)
{
  const unsigned tid = blockIdx.x * 256u + threadIdx.x;   // grid sized to E*32 exactly
  const unsigned e = tid >> 5;
  const unsigned c = (tid & 31u) << 2;
  const int s = src[e];
  const int d = dst[e];
  const float4 v = *(const float4*)(h + (size_t)s * H + c);
  float* p = xn + (size_t)d * H + c;
  atomicAdd(p + 0, v.x);
  atomicAdd(p + 1, v.y);
  atomicAdd(p + 2, v.z);
  atomicAdd(p + 3, v.w);
}

// ---------------------------------------------------------------------------
// Fused GRU layer: per block, 32 nodes (2 M-tiles); per wave, 16 columns of H.
// 12 fp32 WMMA accumulators; every weight B-fragment feeds two WMMAs.
// ---------------------------------------------------------------------------
__device__ __forceinline__ v8f wmma_f32(v2f a, v2f b, v8f c) {
  return __builtin_amdgcn_wmma_f32_16x16x4_f32(false, a, false, b,
                                               (short)0, c, false, false);
}

__global__ __launch_bounds__(256) void ggnn_gru(
    const float* __restrict__ xn, const float* __restrict__ h,
    const float* __restrict__ Wi, const float* __restrict__ Wh,
    const float* __restrict__ bi, const float* __restrict__ bh,
    float* __restrict__ hout)
{
  __shared__ float s_x[32 * H];
  __shared__ float s_h[32 * H];
  const int node0 = blockIdx.x * 32;
  const int tid = threadIdx.x;
  for (int i = tid; i < 32 * H; i += 256) {
    s_x[i] = xn[(size_t)node0 * H + i];
    s_h[i] = h [(size_t)node0 * H + i];
  }
  __syncthreads();

  const int wave = tid >> 5;          // 0..7 -> column tile of H
  const int lane = tid & 31;
  const int half = lane >> 4;         // 0 or 1
  const int n    = lane & 15;         // N index (and A-matrix row index)
  const int jh16 = wave * 16;

  v8f cir[2] = {{}, {}}, ciz[2] = {{}, {}}, cin[2] = {{}, {}};
  v8f chr_[2] = {{}, {}}, chz[2] = {{}, {}}, chn[2] = {{}, {}};

  const float* wi0 = Wi + (size_t)(       jh16 + n) * H;
  const float* wi1 = Wi + (size_t)(H   +  jh16 + n) * H;
  const float* wi2 = Wi + (size_t)(2*H +  jh16 + n) * H;
  const float* wh0 = Wh + (size_t)(       jh16 + n) * H;
  const float* wh1 = Wh + (size_t)(H   +  jh16 + n) * H;
  const float* wh2 = Wh + (size_t)(2*H +  jh16 + n) * H;

  #pragma unroll 4
  for (int k0 = 0; k0 < H; k0 += 4) {
    const int ka = k0 + 2 * half;     // K = 2*half + vgpr  (ISA 16x4 f32 A layout)
    v2f ax0 = *(const v2f*)(&s_x[(     n) * H + ka]);
    v2f ax1 = *(const v2f*)(&s_x[(16 + n) * H + ka]);
    v2f ah0 = *(const v2f*)(&s_h[(     n) * H + ka]);
    v2f ah1 = *(const v2f*)(&s_h[(16 + n) * H + ka]);
    v2f b0 = *(const v2f*)(wi0 + ka);
    v2f b1 = *(const v2f*)(wi1 + ka);
    v2f b2 = *(const v2f*)(wi2 + ka);
    v2f c0 = *(const v2f*)(wh0 + ka);
    v2f c1 = *(const v2f*)(wh1 + ka);
    v2f c2 = *(const v2f*)(wh2 + ka);
    cir[0]  = wmma_f32(ax0, b0, cir[0]);   cir[1]  = wmma_f32(ax1, b0, cir[1]);
    ciz[0]  = wmma_f32(ax0, b1, ciz[0]);   ciz[1]  = wmma_f32(ax1, b1, ciz[1]);
    cin[0]  = wmma_f32(ax0, b2, cin[0]);   cin[1]  = wmma_f32(ax1, b2, cin[1]);
    chr_[0] = wmma_f32(ah0, c0, chr_[0]);  chr_[1] = wmma_f32(ah1, c0, chr_[1]);
    chz[0]  = wmma_f32(ah0, c1, chz[0]);   chz[1]  = wmma_f32(ah1, c1, chz[1]);
    chn[0]  = wmma_f32(ah0, c2, chn[0]);   chn[1]  = wmma_f32(ah1, c2, chn[1]);
  }

  const int colH = jh16 + n;          // 0..127
  const float bir = bi[colH], biz = bi[H + colH], bin_ = bi[2*H + colH];
  const float bhr = bh[colH], bhz = bh[H + colH], bhn  = bh[2*H + colH];

  #pragma unroll
  for (int mt = 0; mt < 2; ++mt) {
    #pragma unroll
    for (int c = 0; c < 8; ++c) {
      const int m = mt * 16 + c + 8 * half;  // C/D layout: M = vgpr + 8*(lane>=16)
      const float gr = (cir[mt][c] + bir) + (chr_[mt][c] + bhr);
      const float gz = (ciz[mt][c] + biz) + (chz[mt][c]  + bhz);
      const float r  = 1.f / (1.f + __expf(-gr));
      const float z  = 1.f / (1.f + __expf(-gz));
      const float gn = (cin[mt][c] + bin_) + r * (chn[mt][c] + bhn);
      const float e2 = __expf(2.f * gn);
      const float nn = (e2 - 1.f) / (e2 + 1.f);   // tanh
      const float ho = s_h[m * H + colH];
      hout[(size_t)(node0 + m) * H + colH] = (1.f - z) * nn + z * ho;
    }
  }
}

// ---------------------------------------------------------------------------
// Final dense + per-block column max (WMMA) over REAL rows only (16-row tiles),
// then global column-max reduce.
// ---------------------------------------------------------------------------
__global__ __launch_bounds__(256) void ggnn_dense_max(
    const float* __restrict__ h, const float* __restrict__ Wd,
    const float* __restrict__ bd, float* __restrict__ partial)
{
  __shared__ float s_h[16 * H];
  const int node0 = blockIdx.x * 16;
  const int tid = threadIdx.x;
  for (int i = tid; i < 16 * H; i += 256)
    s_h[i] = h[(size_t)node0 * H + i];
  __syncthreads();

  const int wave = tid >> 5;
  const int lane = tid & 31;
  const int half = lane >> 4;
  const int n    = lane & 15;
  const int jh16 = wave * 16;

  v8f acc = {};
  const float* wd = Wd + (size_t)(jh16 + n) * H;
  #pragma unroll 4
  for (int k0 = 0; k0 < H; k0 += 4) {
    const int ka = k0 + 2 * half;
    v2f a = *(const v2f*)(&s_h[n * H + ka]);
    v2f b = *(const v2f*)(wd + ka);
    acc = wmma_f32(a, b, acc);
  }

  const float bv = bd[jh16 + n];
  float m = -3.402823466e38f;
  #pragma unroll
  for (int c = 0; c < 8; ++c) m = fmaxf(m, acc[c] + bv);
  m = fmaxf(m, __shfl_xor(m, 16, 32));   // merge the two lane-halves (wave32)
  if (half == 0) partial[(size_t)blockIdx.x * H + jh16 + n] = m;
}

__global__ __launch_bounds__(128) void ggnn_colmax(
    const float* __restrict__ partial, float* __restrict__ out, int nblk)
{
  const int j = threadIdx.x;            // 128 threads, one per output column
  float m = -3.402823466e38f;
  for (int b = 0; b < nblk; ++b) m = fmaxf(m, partial[(size_t)b * H + j]);
  out[j] = m;
}

// ---------------------------------------------------------------------------
extern "C" void kernel_launch(void* const* d_in, const int* in_sizes, int n_in,
                              void* d_out, int out_size, void* d_ws, size_t ws_size,
                              hipStream_t stream) {
  const float* x    = (const float*)d_in[0];
  const int*   ei   = (const int*)d_in[1];
  const float* w_ih = (const float*)d_in[2];   // [L, 3H, H]
  const float* w_hh = (const float*)d_in[3];
  const float* b_ih = (const float*)d_in[4];   // [L, 3H]
  const float* b_hh = (const float*)d_in[5];
  const float* w_d  = (const float*)d_in[6];   // [H, H]
  const float* b_d  = (const float*)d_in[7];   // [H]
  float* out = (float*)d_out;

  const size_t nh  = (size_t)N  * H;           // real rows
  const size_t nhp = (size_t)NP * H;           // padded rows
  float* hA      = (float*)d_ws;
  float* hB      = hA + nhp;
  float* xn      = hB + nhp;
  float* partial = xn + nhp;

  // init hidden state: real rows from x, pad rows zeroed (stay confined to pad)
  hipMemcpyAsync(hA, x, nh * sizeof(float), hipMemcpyDeviceToDevice, stream);
  hipMemsetAsync(hA + nh, 0, (nhp - nh) * sizeof(float), stream);
  hipMemsetAsync(hB + nh, 0, (nhp - nh) * sizeof(float), stream);

  const int* src = ei;
  const int* dst = ei + E;
  float* hc = hA;
  float* hn = hB;

  const int scatter_blocks = (E * 32) / 256;   // 75000, exact
  for (int l = 0; l < 4; ++l) {
    hipMemsetAsync(xn, 0, nhp * sizeof(float), stream);
    ggnn_scatter<<<scatter_blocks, 256, 0, stream>>>(hc, src, dst, xn);
    ggnn_gru<<<GBLK, 256, 0, stream>>>(
        xn, hc,
        w_ih + (size_t)l * H3 * H, w_hh + (size_t)l * H3 * H,
        b_ih + (size_t)l * H3,     b_hh + (size_t)l * H3,
        hn);
    float* t = hc; hc = hn; hn = t;
  }

  ggnn_dense_max<<<MBLK, 256, 0, stream>>>(hc, w_d, b_d, partial);
  ggnn_colmax<<<1, 128, 0, stream>>>(partial, out, MBLK);
}